// LBilin_26199300505666
// MI455X (gfx1250) — compile-verified
//
#include <hip/hip_runtime.h>
#include <hip/hip_bf16.h>

typedef __attribute__((ext_vector_type(2))) float v2f;
typedef __attribute__((ext_vector_type(8))) float v8f;

#define X_SITES   65536
#define WAVES_PB  8
#define U_FLOATS  (X_SITES * 72)   // 4,718,592

// ---------------------------------------------------------------------------
// Kernel 1: pass-through copy of u (first element of the output tuple)
// ---------------------------------------------------------------------------
__global__ __launch_bounds__(256)
void lbilin_copy_u(const float4* __restrict__ src, float4* __restrict__ dst, int n4) {
    int i = blockIdx.x * blockDim.x + threadIdx.x;
    if (i < n4) dst[i] = src[i];
}

// ---------------------------------------------------------------------------
// Kernel 2: per-site bilinear + channel mix via V_WMMA_F32_16X16X4_F32.
//   Stage 1: T[72x18] = WT[72x9] @ W2f[9x18]      (WT = reshaped weight)
//   Stage 2: Out[6x24] = A2[6x54] @ B2[54x24]     (complex via 2x2 real embed)
// One wave32 per site, 8 waves per 256-thread block.
// w1/w2 site tiles staged with GLOBAL_LOAD_ASYNC_TO_LDS_B128 (ASYNCcnt).
// ---------------------------------------------------------------------------
__global__ __launch_bounds__(256)
void lbilin_wmma(const float* __restrict__ w1g, const float* __restrict__ w2g,
                 const float* __restrict__ wg,  float* __restrict__ outw) {
    __shared__ float s_weight[648];          // weight[8][9][9]
    __shared__ float s_w1e[WAVES_PB][162];   // w1e[9][3][3][2] per wave's site
    __shared__ float s_w2e[WAVES_PB][162];   // w2e[9][3][3][2]
    __shared__ float s_T[WAVES_PB][1296];    // T[72][18]

    const int tid  = threadIdx.x;
    const int lane = tid & 31;
    const int wv   = tid >> 5;
    const int l16  = lane & 15;
    const int hi   = lane >> 4;              // 0: lanes 0-15, 1: lanes 16-31
    const int site = blockIdx.x * WAVES_PB + wv;

    // --- async-copy this wave's w1/w2 site tiles (576 B each) into LDS ---
    {
        const unsigned long long ga1 = (unsigned long long)(w1g + (size_t)site * 144);
        const unsigned long long ga2 = (unsigned long long)(w2g + (size_t)site * 144);
        const unsigned int la1 = (unsigned int)(uintptr_t)(&s_w1e[wv][0]);
        const unsigned int la2 = (unsigned int)(uintptr_t)(&s_w2e[wv][0]);

        unsigned long long gaA = ga1 + (unsigned)lane * 16u;
        unsigned int       laA = la1 + (unsigned)lane * 16u;
        asm volatile("global_load_async_to_lds_b128 %0, %1, off"
                     :: "v"(laA), "v"(gaA) : "memory");
        unsigned long long gaB = ga2 + (unsigned)lane * 16u;
        unsigned int       laB = la2 + (unsigned)lane * 16u;
        asm volatile("global_load_async_to_lds_b128 %0, %1, off"
                     :: "v"(laB), "v"(gaB) : "memory");
        if (lane < 8) {   // tail: 64 B per array; lanes 0-3 -> w1, 4-7 -> w2
            const int idx = lane & 3;
            const bool isw2 = lane >= 4;
            unsigned long long gaT = (isw2 ? ga2 : ga1) + 512u + (unsigned)idx * 16u;
            unsigned int       laT = (isw2 ? la2 : la1) + 512u + (unsigned)idx * 16u;
            asm volatile("global_load_async_to_lds_b128 %0, %1, off"
                         :: "v"(laT), "v"(gaT) : "memory");
        }
    }

    // --- weight into LDS (block-wide, overlaps with async transfers) ---
    for (int i = tid; i < 648; i += 256) s_weight[i] = wg[i];

    // --- identity channel (ch 8) ---
    for (int i = lane; i < 18; i += 32) {
        int jk = i >> 1, c = i & 1;
        float idv = (c == 0 && (jk == 0 || jk == 4 || jk == 8)) ? 1.0f : 0.0f;
        s_w1e[wv][144 + i] = idv;
        s_w2e[wv][144 + i] = idv;
    }

    asm volatile("s_wait_asynccnt 0x0" ::: "memory");
    __syncthreads();

    const float* w1e = s_w1e[wv];
    const float* w2e = s_w2e[wv];
    float*       T   = s_T[wv];

    // =======================================================================
    // Stage 1: T[(v*8+u)][j*6+k*2+c] = sum_w weight[u][v][w] * w2e[w][j,k,c]
    //   GEMM M=72 (pad 80), K=9 (pad 12), N=18 (pad 32)
    // =======================================================================
    // B fragments depend only on (nt, kt): build once.
    v2f bfr[2][3];
    #pragma unroll
    for (int nt = 0; nt < 2; ++nt) {
        const int n   = nt * 16 + l16;
        const int nCl = (n < 18) ? n : 0;
        #pragma unroll
        for (int kt = 0; kt < 3; ++kt) {
            const int k0 = kt * 4 + hi * 2;
            #pragma unroll
            for (int t = 0; t < 2; ++t) {
                const int k  = k0 + t;
                const int kc = (k < 9) ? k : 0;
                float bv = w2e[kc * 18 + nCl];
                bfr[nt][kt][t] = (n < 18 && k < 9) ? bv : 0.0f;
            }
        }
    }
    #pragma unroll
    for (int mt = 0; mt < 5; ++mt) {
        const int m   = mt * 16 + l16;           // WT row
        const int mCl = (m < 72) ? m : 0;
        const int vch = mCl >> 3, uo = mCl & 7;
        v2f afr[3];                              // A fragments for this mt
        #pragma unroll
        for (int kt = 0; kt < 3; ++kt) {
            const int k0 = kt * 4 + hi * 2;
            #pragma unroll
            for (int t = 0; t < 2; ++t) {
                const int k  = k0 + t;
                const int kc = (k < 9) ? k : 0;
                float av = s_weight[uo * 81 + vch * 9 + kc];
                afr[kt][t] = (m < 72 && k < 9) ? av : 0.0f;
            }
        }
        #pragma unroll
        for (int nt = 0; nt < 2; ++nt) {
            v8f acc = {};
            #pragma unroll
            for (int kt = 0; kt < 3; ++kt) {
                acc = __builtin_amdgcn_wmma_f32_16x16x4_f32(
                        false, afr[kt], false, bfr[nt][kt], (short)0, acc,
                        false, false);
            }
            #pragma unroll
            for (int r = 0; r < 8; ++r) {
                const int R = mt * 16 + hi * 8 + r;
                const int C = nt * 16 + l16;
                if (R < 72 && C < 18) T[R * 18 + C] = acc[r];
            }
        }
    }
    __syncthreads();

    // =======================================================================
    // Stage 2: Out[6x24] = A2[6x54] @ B2[54x24]
    //   A2 rows (cr,i): [[w1R, -w1I],[w1I, w1R]], cols (cc,v,j)
    //   B2 rows (cc,v,j) -> T[(v,u)][j,k,cc], cols (u,k)
    //   K=54 (pad 56, 14 steps), N=24 (2 tiles), M=6 (pad 16)
    // =======================================================================
    const int out_base = site * 144;
    const int cr = l16 / 3, i2 = l16 % 3;        // A2 row decomp (valid l16<6)

    // A fragments depend only on kt: build once.
    v2f a2[14];
    #pragma unroll
    for (int kt = 0; kt < 14; ++kt) {
        const int k0 = kt * 4 + hi * 2;
        #pragma unroll
        for (int t = 0; t < 2; ++t) {
            const int k   = k0 + t;              // K dim: (cc, v, j)
            const int kc  = (k < 54) ? k : 0;
            const int cc  = kc / 27, rem = kc % 27;
            const int vch = rem / 3, j = rem % 3;
            const float Rr = w1e[vch * 18 + (i2 * 3 + j) * 2 + 0];
            const float Ii = w1e[vch * 18 + (i2 * 3 + j) * 2 + 1];
            float av = (cc == 0) ? ((cr == 0) ? Rr : Ii)
                                 : ((cr == 0) ? -Ii : Rr);
            a2[kt][t] = (l16 < 6 && k < 54) ? av : 0.0f;
        }
    }
    #pragma unroll
    for (int nt = 0; nt < 2; ++nt) {
        v8f acc = {};
        const int n   = nt * 16 + l16;           // B2 col
        const int nCl = (n < 24) ? n : 0;
        const int uo  = nCl / 3, k2 = nCl % 3;
        #pragma unroll
        for (int kt = 0; kt < 14; ++kt) {
            const int k0 = kt * 4 + hi * 2;
            v2f b;
            #pragma unroll
            for (int t = 0; t < 2; ++t) {
                const int k   = k0 + t;
                const int kc  = (k < 54) ? k : 0;
                const int cc  = kc / 27, rem = kc % 27;
                const int vch = rem / 3, j = rem % 3;
                float bv = T[(vch * 8 + uo) * 18 + j * 6 + k2 * 2 + cc];
                b[t] = (n < 24 && k < 54) ? bv : 0.0f;
            }
            acc = __builtin_amdgcn_wmma_f32_16x16x4_f32(
                    false, a2[kt], false, b, (short)0, acc, false, false);
        }
        // Scatter D: row = hi*8 + r (valid rows 0..5), col = (u,k)
        #pragma unroll
        for (int r = 0; r < 8; ++r) {
            const int row = hi * 8 + r;
            const int col = nt * 16 + l16;
            if (row < 6 && col < 24) {
                const int crr = row / 3, ii = row % 3;
                const int uu  = col / 3, kk = col % 3;
                outw[out_base + uu * 18 + ii * 6 + kk * 2 + crr] = acc[r];
            }
        }
    }
}

// ---------------------------------------------------------------------------
// Launch
// ---------------------------------------------------------------------------
extern "C" void kernel_launch(void* const* d_in, const int* in_sizes, int n_in,
                              void* d_out, int out_size, void* d_ws, size_t ws_size,
                              hipStream_t stream) {
    const float* u  = (const float*)d_in[0];
    const float* w1 = (const float*)d_in[1];
    const float* w2 = (const float*)d_in[2];
    const float* wg = (const float*)d_in[3];
    float* out = (float*)d_out;

    // Output = concat(u, w): copy u first
    const int n4 = U_FLOATS / 4;               // 1,179,648 float4
    lbilin_copy_u<<<(n4 + 255) / 256, 256, 0, stream>>>(
        (const float4*)u, (float4*)out, n4);

    // Bilinear + channel mix: 65536 sites, 8 waves/block
    lbilin_wmma<<<X_SITES / WAVES_PB, 256, 0, stream>>>(
        w1, w2, wg, out + U_FLOATS);
}